// DHBR_65481071395086
// MI455X (gfx1250) — compile-verified
//
#include <hip/hip_runtime.h>

#define N_NODES 200000
#define NPART   256
#define CHUNK   ((N_NODES + NPART - 1) / NPART)   // 782

typedef float v2f __attribute__((ext_vector_type(2)));
typedef float v8f __attribute__((ext_vector_type(8)));

__device__ __forceinline__ float leaky(float x) {
    return x >= 0.0f ? x : 0.05f * x;
}

// ---------------------------------------------------------------------------
// CDNA5 async global->LDS copy (ASYNCcnt-tracked), 16 bytes per lane.
// Generic pointers to __shared__ carry the LDS offset in their low 32 bits
// (flat aperture: offset is addr bits 31..0).
// ---------------------------------------------------------------------------
__device__ __forceinline__ void async_copy16(const float* gsrc, float* lds_dst) {
    unsigned lds_off = (unsigned)(size_t)lds_dst;
    asm volatile("global_load_async_to_lds_b128 %0, %1, off"
                 :: "v"(lds_off), "v"(gsrc) : "memory");
}
__device__ __forceinline__ void wait_async0() {
    asm volatile("s_wait_asynccnt 0" ::: "memory");
}

// ---------------------------------------------------------------------------
// Inner WMMA tile: acc[16(H)x128(E)] += lds_a(64x128)^T-frag x lds_b(64x128)
// Wave32 f32 16x16x4 operand layouts per ISA 7.12.2.
// ---------------------------------------------------------------------------
__device__ __forceinline__ void mma_tile64(const float* lds_a, const float* lds_b,
                                           v8f acc[8], int h0, int lo, int hi) {
    #pragma unroll 4
    for (int k = 0; k < 64; k += 4) {
        const int kr = k + hi * 2;
        v2f a;   // A[M=h0+lo][K] = lds_a[K][h0+lo] (row read, conflict-free)
        a.x = lds_a[(kr    ) * 128 + h0 + lo];
        a.y = lds_a[(kr + 1) * 128 + h0 + lo];
        #pragma unroll
        for (int e = 0; e < 8; ++e) {
            v2f b;
            b.x = lds_b[(kr    ) * 128 + e * 16 + lo];
            b.y = lds_b[(kr + 1) * 128 + e * 16 + lo];
            acc[e] = __builtin_amdgcn_wmma_f32_16x16x4_f32(
                false, a, false, b, (short)0, acc[e], false, false);
        }
    }
}

// ---------------------------------------------------------------------------
// Stage A: partial[p] = adj[rows_p, :]^T @ feature[rows_p, :]   (128x128 f32)
// Split-K over N across NPART blocks; double-buffered async tiles.
// ---------------------------------------------------------------------------
__global__ __launch_bounds__(256) void stageA_kernel(
    const float* __restrict__ adj, const float* __restrict__ feat,
    float* __restrict__ partial)
{
    __shared__ float lds_adj[2][64 * 128];   // 2 x 32 KB
    __shared__ float lds_ft [2][64 * 128];   // 2 x 32 KB  (128 KB total)

    const int  p    = blockIdx.x;
    const long row0 = (long)p * CHUNK;
    long row1 = row0 + (long)CHUNK;
    if (row1 > (long)N_NODES) row1 = (long)N_NODES;

    const int tid  = threadIdx.x;
    const int lane = tid & 31;
    const int lo   = lane & 15;
    const int hi   = lane >> 4;
    const int h0   = (tid >> 5) * 16;

    const int nfull = (int)((row1 - row0) >> 6);   // full 64-row tiles

    v8f acc[8] = {};

    // Prefetch tile 0
    if (nfull > 0) {
        #pragma unroll
        for (int i = tid; i < 2048; i += 256) {
            long r  = row0 + (i >> 5);
            int  c4 = (i & 31) * 4;
            async_copy16(adj  + r * 128 + c4, &lds_adj[0][i * 4]);
            async_copy16(feat + r * 128 + c4, &lds_ft [0][i * 4]);
        }
    }

    for (int c = 0; c < nfull; ++c) {
        wait_async0();          // this wave's tile-c data has landed
        __syncthreads();        // every wave's tile-c data has landed
        if (c + 1 < nfull) {    // prefetch tile c+1 into the other buffer
            const long base = row0 + (long)(c + 1) * 64;
            const int  nb   = (c + 1) & 1;
            #pragma unroll
            for (int i = tid; i < 2048; i += 256) {
                long r  = base + (i >> 5);
                int  c4 = (i & 31) * 4;
                async_copy16(adj  + r * 128 + c4, &lds_adj[nb][i * 4]);
                async_copy16(feat + r * 128 + c4, &lds_ft [nb][i * 4]);
            }
        }
        mma_tile64(lds_adj[c & 1], lds_ft[c & 1], acc, h0, lo, hi);
        __syncthreads();        // reads done before this buffer is refilled
    }

    // Tail (< 64 rows): synchronous zero-padded fill into buffer 0
    const long tbase = row0 + (long)nfull * 64;
    if (tbase < row1) {
        for (int i = tid; i < 2048; i += 256) {
            long r = tbase + (i >> 5);
            float4 va, vf;
            if (r < row1) {
                va = ((const float4*)(adj  + r * 128))[i & 31];
                vf = ((const float4*)(feat + r * 128))[i & 31];
            } else {
                va = make_float4(0.f, 0.f, 0.f, 0.f);
                vf = va;
            }
            ((float4*)lds_adj[0])[i] = va;
            ((float4*)lds_ft [0])[i] = vf;
        }
        __syncthreads();
        mma_tile64(lds_adj[0], lds_ft[0], acc, h0, lo, hi);
    }

    float* outp = partial + (long)p * (128 * 128);
    #pragma unroll
    for (int e = 0; e < 8; ++e)
        #pragma unroll
        for (int v = 0; v < 8; ++v)
            outp[(h0 + v + 8 * hi) * 128 + e * 16 + lo] = acc[e][v];
}

// ---------------------------------------------------------------------------
// Deterministic partial reduction + leaky -> f1 [128x128]
// ---------------------------------------------------------------------------
__global__ __launch_bounds__(256) void reduce_kernel(
    const float* __restrict__ partial, float* __restrict__ f1)
{
    const int idx = blockIdx.x * 256 + threadIdx.x;   // 0..16383
    float s = 0.0f;
    #pragma unroll 8
    for (int p = 0; p < NPART; ++p)
        s += partial[(long)p * (128 * 128) + idx];
    f1[idx] = leaky(s);
}

// ---------------------------------------------------------------------------
// Chain stage: fout = leaky(w @ fin) + fin     (all 128x128, single block)
// ---------------------------------------------------------------------------
__global__ __launch_bounds__(256) void chain_kernel(
    const float* __restrict__ w, const float* __restrict__ fin,
    float* __restrict__ fout)
{
    __shared__ float lds_w[128 * 132];   // padded stride (column A-reads)
    __shared__ float lds_f[128 * 128];

    const int tid = threadIdx.x;
    for (int i = tid; i < (128 * 128) / 4; i += 256) {
        int r = i >> 5;
        int c = (i & 31) * 4;
        *(float4*)(lds_w + r * 132 + c) = ((const float4*)w)[i];
        ((float4*)lds_f)[i]             = ((const float4*)fin)[i];
    }
    __syncthreads();

    const int lane = tid & 31, lo = lane & 15, hi = lane >> 4;
    const int h0 = (tid >> 5) * 16;

    v8f acc[8] = {};
    #pragma unroll 4
    for (int k = 0; k < 128; k += 4) {
        const int kr = k + hi * 2;
        v2f a;
        a.x = lds_w[(h0 + lo) * 132 + kr    ];
        a.y = lds_w[(h0 + lo) * 132 + kr + 1];
        #pragma unroll
        for (int e = 0; e < 8; ++e) {
            v2f b;
            b.x = lds_f[(kr    ) * 128 + e * 16 + lo];
            b.y = lds_f[(kr + 1) * 128 + e * 16 + lo];
            acc[e] = __builtin_amdgcn_wmma_f32_16x16x4_f32(
                false, a, false, b, (short)0, acc[e], false, false);
        }
    }

    #pragma unroll
    for (int e = 0; e < 8; ++e)
        #pragma unroll
        for (int v = 0; v < 8; ++v) {
            int r = h0 + v + 8 * hi, c = e * 16 + lo;
            fout[r * 128 + c] = leaky(acc[e][v]) + lds_f[r * 128 + c];
        }
}

// ---------------------------------------------------------------------------
// Stage C: out = leaky(adj @ f4)     ([N,128] x [128,128])
// Each block handles a 128-row tile of adj; tiles loaded via async-to-LDS.
// ---------------------------------------------------------------------------
__global__ __launch_bounds__(256) void stageC_kernel(
    const float* __restrict__ adj, const float* __restrict__ f4,
    float* __restrict__ out)
{
    __shared__ float lds_a[128 * 132];   // padded stride (column A-reads)
    __shared__ float lds_f[128 * 128];

    const long m0  = (long)blockIdx.x * 128;
    const int  tid = threadIdx.x;

    for (int i = tid; i < 4096; i += 256) {
        int  r  = i >> 5;
        int  c4 = (i & 31) * 4;
        long gr = m0 + r;
        float* dsta = lds_a + r * 132 + c4;
        if (gr < (long)N_NODES) {
            async_copy16(adj + gr * 128 + c4, dsta);
        } else {
            *(float4*)dsta = make_float4(0.f, 0.f, 0.f, 0.f);
        }
        async_copy16(f4 + (size_t)i * 4, lds_f + i * 4);
    }
    wait_async0();
    __syncthreads();

    const int lane = tid & 31, lo = lane & 15, hi = lane >> 4;
    const int r0 = (tid >> 5) * 16;

    v8f acc[8] = {};
    #pragma unroll 4
    for (int k = 0; k < 128; k += 4) {
        const int kr = k + hi * 2;
        v2f a;
        a.x = lds_a[(r0 + lo) * 132 + kr    ];
        a.y = lds_a[(r0 + lo) * 132 + kr + 1];
        #pragma unroll
        for (int e = 0; e < 8; ++e) {
            v2f b;
            b.x = lds_f[(kr    ) * 128 + e * 16 + lo];
            b.y = lds_f[(kr + 1) * 128 + e * 16 + lo];
            acc[e] = __builtin_amdgcn_wmma_f32_16x16x4_f32(
                false, a, false, b, (short)0, acc[e], false, false);
        }
    }

    #pragma unroll
    for (int e = 0; e < 8; ++e)
        #pragma unroll
        for (int v = 0; v < 8; ++v) {
            long row = m0 + r0 + v + 8 * hi;
            if (row < (long)N_NODES)
                out[row * 128 + e * 16 + lo] = leaky(acc[e][v]);
        }
}

// ---------------------------------------------------------------------------
extern "C" void kernel_launch(void* const* d_in, const int* in_sizes, int n_in,
                              void* d_out, int out_size, void* d_ws, size_t ws_size,
                              hipStream_t stream)
{
    const float* feature = (const float*)d_in[0];   // [N,128]
    const float* adj     = (const float*)d_in[1];   // [N,128]
    const float* w1      = (const float*)d_in[2];   // [128,128]
    const float* w2      = (const float*)d_in[3];
    const float* w3      = (const float*)d_in[4];
    float* out = (float*)d_out;

    float* ws      = (float*)d_ws;
    float* partial = ws;                                   // NPART*128*128 f32
    float* f1      = ws + (size_t)NPART * 128 * 128;
    float* f2      = f1 + 128 * 128;
    float* f3      = f2 + 128 * 128;
    float* f4      = f3 + 128 * 128;

    stageA_kernel<<<NPART, 256, 0, stream>>>(adj, feature, partial);
    reduce_kernel<<<(128 * 128) / 256, 256, 0, stream>>>(partial, f1);
    chain_kernel<<<1, 256, 0, stream>>>(w1, f1, f2);
    chain_kernel<<<1, 256, 0, stream>>>(w2, f2, f3);
    chain_kernel<<<1, 256, 0, stream>>>(w3, f3, f4);
    stageC_kernel<<<(N_NODES + 127) / 128, 256, 0, stream>>>(adj, f4, out);
}